// TopKActivation_86560770883884
// MI455X (gfx1250) — compile-verified
//
#include <hip/hip_runtime.h>
#include <stdint.h>

// Top-K (K=64) per row with ReLU, scatter into zeros.
// x: [ROWS, 24576] fp32 -> out: same shape.
// 1 block per row; async-copy row into LDS (gfx1250 async-to-LDS + asynccnt),
// radix-select the 64th-largest key with per-wave LDS histograms + parallel
// suffix scan, then one vectorized non-temporal pass writes the output.

#define N_COLS   24576
#define K_KEEP   64
#define BLOCK    512
#define CHUNKS   (N_COLS / 4)        // 6144 float4 chunks per row
#define ITERS    (CHUNKS / BLOCK)    // 12, exact
#define NWAVE    (BLOCK / 32)        // 16 wave32s
#define HSTRIDE  257                 // 256 bins + 1 pad -> per-wave bank skew

typedef float vfloat4 __attribute__((ext_vector_type(4)));

// Monotone mapping: float total order == unsigned order of key.
__device__ __forceinline__ unsigned f2key(float f) {
    unsigned u = __float_as_uint(f);
    return (u & 0x80000000u) ? ~u : (u | 0x80000000u);
}

__global__ __launch_bounds__(BLOCK)
void topk_relu_scatter_kernel(const float* __restrict__ x,
                              float* __restrict__ out) {
    __shared__ float    srow[N_COLS];            // 96 KB row tile
    __shared__ unsigned whist[NWAVE * HSTRIDE];  // per-wave histograms (16.1 KB)
    __shared__ unsigned sufx[257];               // suffix-scan scratch
    __shared__ unsigned s_prefix;
    __shared__ int      s_remaining;
    __shared__ int      s_tiectr;

    const int tid  = threadIdx.x;
    const int wave = tid >> 5;
    const int row  = blockIdx.x;
    const float* rowp = x   + (size_t)row * N_COLS;
    float*       orow = out + (size_t)row * N_COLS;

    // ---------------- async copy: global row -> LDS ----------------
    // IOFFSET applies to BOTH the global address and the LDS destination,
    // so one VGPR pair + one LDS-addr VGPR serve all 12 chunks per thread.
    {
        unsigned     loff = (unsigned)(uintptr_t)(&srow[0]) + (unsigned)tid * 16u;
        const float* g    = rowp + (size_t)tid * 4;
        #pragma unroll
        for (int i = 0; i < ITERS; ++i) {
            asm volatile("global_load_async_to_lds_b128 %0, %1, off offset:%2 th:TH_LOAD_NT"
                         :: "v"(loff), "v"(g), "i"(i * BLOCK * 16) : "memory");
        }
    }
    asm volatile("s_wait_asynccnt 0x0" ::: "memory");

    if (tid == 0) { s_prefix = 0u; s_remaining = K_KEEP; }
    __syncthreads();

    // ---------------- radix select: key of the K-th largest ----------------
    for (int pass = 3; pass >= 0; --pass) {
        // zero per-wave histograms
        for (int i = tid; i < NWAVE * HSTRIDE; i += BLOCK) whist[i] = 0u;
        __syncthreads();
        const unsigned prefix = s_prefix;
        const int      shift  = pass * 8;
        unsigned* myh = &whist[wave * HSTRIDE];

        #pragma unroll
        for (int i = 0; i < ITERS; ++i) {
            const int c = tid + i * BLOCK;
            vfloat4 v = reinterpret_cast<const vfloat4*>(srow)[c];
            #pragma unroll
            for (int j = 0; j < 4; ++j) {
                unsigned key = f2key(v[j]);
                bool in = (pass == 3) || ((key >> (shift + 8)) == prefix);
                if (in) atomicAdd(&myh[(key >> shift) & 0xFFu], 1u);
            }
        }
        __syncthreads();

        // reduce 16 per-wave histograms into sufx[0..255]
        if (tid < 256) {
            unsigned s = 0;
            #pragma unroll
            for (int w = 0; w < NWAVE; ++w) s += whist[w * HSTRIDE + tid];
            sufx[tid] = s;
        }
        __syncthreads();

        // parallel suffix sum: sufx[t] = count of elements with bin >= t
        #pragma unroll
        for (int d = 1; d < 256; d <<= 1) {
            unsigned v = 0;
            if (tid < 256) {
                v = sufx[tid];
                if (tid + d < 256) v += sufx[tid + d];
            }
            __syncthreads();
            if (tid < 256) sufx[tid] = v;
            __syncthreads();
        }

        // pick bin b: count(>b) < rem <= count(>=b)
        const unsigned rem = (unsigned)s_remaining;
        const unsigned pfx = s_prefix;
        __syncthreads();
        if (tid < 256) {
            unsigned ge = sufx[tid];
            unsigned gt = (tid == 255) ? 0u : sufx[tid + 1];
            if (ge >= rem && gt < rem) {
                s_prefix    = (pfx << 8) | (unsigned)tid;
                s_remaining = (int)(rem - gt);
            }
        }
        __syncthreads();
    }

    const unsigned T          = s_prefix;     // key of K-th largest
    const int      tie_budget = s_remaining;  // # of ==T elements to keep
    if (tid == 0) s_tiectr = 0;
    __syncthreads();

    // ---------------- output: LDS -> global, b128 non-temporal ----------------
    #pragma unroll
    for (int i = 0; i < ITERS; ++i) {
        const int c = tid + i * BLOCK;
        vfloat4 v = reinterpret_cast<const vfloat4*>(srow)[c];
        vfloat4 r;
        #pragma unroll
        for (int j = 0; j < 4; ++j) {
            unsigned key  = f2key(v[j]);
            float    kept = v[j] > 0.0f ? v[j] : 0.0f;   // ReLU of kept value
            if (key > T) {
                r[j] = kept;
            } else if (key == T) {
                int pos = atomicAdd(&s_tiectr, 1);
                r[j] = (pos < tie_budget) ? kept : 0.0f;
            } else {
                r[j] = 0.0f;
            }
        }
        __builtin_nontemporal_store(r, &reinterpret_cast<vfloat4*>(orow)[c]);
    }
}

extern "C" void kernel_launch(void* const* d_in, const int* in_sizes, int n_in,
                              void* d_out, int out_size, void* d_ws, size_t ws_size,
                              hipStream_t stream) {
    const float* x   = (const float*)d_in[0];
    float*       out = (float*)d_out;
    const int rows = in_sizes[0] / N_COLS;   // 8192
    hipLaunchKernelGGL(topk_relu_scatter_kernel,
                       dim3(rows), dim3(BLOCK), 0, stream, x, out);
}